// PaiNNLayer_29850022707541
// MI455X (gfx1250) — compile-verified
//
#include <hip/hip_runtime.h>

#define EPSF 0.70710678118654752440f  // 1/sqrt(2) layer residual scaling
#define HDIM 128

typedef float v2f __attribute__((ext_vector_type(2)));
typedef float v4f __attribute__((ext_vector_type(4)));
typedef float v8f __attribute__((ext_vector_type(8)));

// ---------------------------------------------------------------------------
// fp32 WMMA helpers (V_WMMA_F32_16X16X4_F32)
//  A 16x4 layout : lanes 0-15 hold {K=0,K=1}, lanes 16-31 hold {K=2,K=3}, M=lane&15
//  B 4x16 layout : mirrored, N=lane&15
//  C/D 16x16     : vgpr v -> row v (lanes 0-15) / row v+8 (lanes 16-31), col=lane&15
// ---------------------------------------------------------------------------
__device__ __forceinline__ v8f wmma_k(const float* __restrict__ A, int lda,
                                      const float* __restrict__ B, int ldb,
                                      int K, int lane, v8f acc) {
  const int half2 = (lane >> 4) << 1;  // 0 or 2
  const int l = lane & 15;
  for (int k = 0; k < K; k += 4) {
    v2f a, b;
    a[0] = A[l * lda + k + half2 + 0];
    a[1] = A[l * lda + k + half2 + 1];
    b[0] = B[(k + half2 + 0) * ldb + l];
    b[1] = B[(k + half2 + 1) * ldb + l];
    acc = __builtin_amdgcn_wmma_f32_16x16x4_f32(false, a, false, b,
                                                (short)0, acc, false, false);
  }
  return acc;
}

__device__ __forceinline__ void store_tile(float* C, int ldc, v8f d, int lane) {
  const int row0 = (lane >> 4) << 3;  // 0 or 8
  const int l = lane & 15;
#pragma unroll
  for (int vg = 0; vg < 8; ++vg) C[(row0 + vg) * ldc + l] = d[vg];
}

__device__ __forceinline__ float silu(float x) {
  return x / (1.0f + __expf(-x));
}

// Native fp32 atomic add (avoid CAS-loop lowering); device scope for L2 RMW.
__device__ __forceinline__ void gatomic_add_f32(float* p, float v) {
  asm volatile("global_atomic_add_f32 %0, %1, off scope:SCOPE_DEV"
               :: "v"(p), "v"(v) : "memory");
}

// ---------------------------------------------------------------------------
// Kernel 1: x = silu(s @ Wi1 + bi1) @ Wi2 + bi2      (N,128) -> (N,384)
// 256 threads = 8 waves, 16 nodes per block.
// ---------------------------------------------------------------------------
__global__ void painn_node_mlp(const float* __restrict__ s,
                               const float* __restrict__ Wi1,
                               const float* __restrict__ bi1,
                               const float* __restrict__ Wi2,
                               const float* __restrict__ bi2,
                               float* __restrict__ x, int N) {
  __shared__ float sA[16 * HDIM];
  __shared__ float hB[16 * HDIM];
  const int block0 = blockIdx.x * 16;
  const int tid = threadIdx.x;
  const int wave = tid >> 5;
  const int lane = tid & 31;

  for (int i = tid; i < 16 * HDIM; i += 256) {
    int nl = i >> 7, h = i & 127;
    int node = block0 + nl;
    sA[i] = (node < N) ? s[(size_t)node * HDIM + h] : 0.0f;
  }
  __syncthreads();

  // layer 1: 128 output cols = 8 tiles, one per wave
  {
    const int n0 = wave * 16;
    v8f acc = {};
    acc = wmma_k(sA, HDIM, Wi1 + n0, HDIM, HDIM, lane, acc);
    const int row0 = (lane >> 4) << 3;
    const int l = lane & 15;
    const float bias = bi1[n0 + l];
#pragma unroll
    for (int vg = 0; vg < 8; ++vg)
      hB[(row0 + vg) * HDIM + n0 + l] = silu(acc[vg] + bias);
  }
  __syncthreads();

  // layer 2: 384 output cols = 24 tiles, 3 per wave
  for (int t = wave; t < 24; t += 8) {
    const int n0 = t * 16;
    v8f acc = {};
    acc = wmma_k(hB, HDIM, Wi2 + n0, 3 * HDIM, HDIM, lane, acc);
    const int row0 = (lane >> 4) << 3;
    const int l = lane & 15;
    const float bias = bi2[n0 + l];
#pragma unroll
    for (int vg = 0; vg < 8; ++vg) {
      int node = block0 + row0 + vg;
      if (node < N) x[(size_t)node * 384 + n0 + l] = acc[vg] + bias;
    }
  }
}

// ---------------------------------------------------------------------------
// Kernel 2: per-edge message + scatter-add.  One wave per edge; lane owns 4
// channels (v4f = 16B accesses).  Wij/dir stream once -> non-temporal loads
// (keep L2 for the reused x/v gather tables and ds/dv accumulators).
// ---------------------------------------------------------------------------
__global__ void painn_edge(const float* __restrict__ Wij,
                           const float* __restrict__ dir,
                           const int* __restrict__ senders,
                           const int* __restrict__ receivers,
                           const float* __restrict__ x,
                           const float* __restrict__ v,
                           float* __restrict__ ds,
                           float* __restrict__ dv, int E) {
  const int e = blockIdx.x * 8 + (threadIdx.x >> 5);
  if (e >= E) return;
  const int lane = threadIdx.x & 31;
  const int r = receivers[e];
  const int snd = senders[e];

  const v4f* W4 = (const v4f*)(Wij + (size_t)e * 384);
  const v4f* X4 = (const v4f*)(x + (size_t)r * 384);
  const v4f* V4 = (const v4f*)(v + (size_t)r * 384);

  // one-shot stream: non-temporal
  v4f w0 = __builtin_nontemporal_load(W4 + lane);
  v4f w1 = __builtin_nontemporal_load(W4 + 32 + lane);
  v4f w2 = __builtin_nontemporal_load(W4 + 64 + lane);
  // reused gather tables: regular temporal (L2-resident)
  v4f x0 = X4[lane], x1 = X4[32 + lane], x2 = X4[64 + lane];

  v4f dse = w0 * x0;
  v4f m1  = w1 * x1;
  v4f m2  = w2 * x2;

  const float d0 = __builtin_nontemporal_load(dir + (size_t)e * 3 + 0);
  const float d1 = __builtin_nontemporal_load(dir + (size_t)e * 3 + 1);
  const float d2 = __builtin_nontemporal_load(dir + (size_t)e * 3 + 2);

  float* dsp = ds + (size_t)snd * 128 + lane * 4;
  gatomic_add_f32(dsp + 0, dse.x);
  gatomic_add_f32(dsp + 1, dse.y);
  gatomic_add_f32(dsp + 2, dse.z);
  gatomic_add_f32(dsp + 3, dse.w);

  float* dvp = dv + (size_t)snd * 384 + lane * 4;
#pragma unroll
  for (int d = 0; d < 3; ++d) {
    const float dd = (d == 0) ? d0 : ((d == 1) ? d1 : d2);
    v4f vj = V4[d * 32 + lane];
    v4f dve = m1 * dd + m2 * vj;
    float* p = dvp + d * 128;
    gatomic_add_f32(p + 0, dve.x);
    gatomic_add_f32(p + 1, dve.y);
    gatomic_add_f32(p + 2, dve.z);
    gatomic_add_f32(p + 3, dve.w);
  }
}

// ---------------------------------------------------------------------------
// Kernel 3: fused intra-particle update.  16 nodes/block, 8 waves,
// 128 KB dynamic LDS (s1|v1|wf|vnorm|dot|h2|u).  WGP(320KB) holds 2 groups.
// ---------------------------------------------------------------------------
__global__ void painn_update(const float* __restrict__ s,
                             const float* __restrict__ v,
                             const float* __restrict__ ds,
                             const float* __restrict__ dv,
                             const float* __restrict__ Wvm,
                             const float* __restrict__ Wm1,
                             const float* __restrict__ bm1,
                             const float* __restrict__ Wm2,
                             const float* __restrict__ bm2,
                             float* __restrict__ out, int N) {
  extern __shared__ float lds[];
  float* s1    = lds;            // 16x128              (2048)
  float* v1    = s1 + 2048;      // 48x128 row=d*16+nl  (6144)
  float* wf    = v1 + 6144;      // 48x256 v@Wvm        (12288)
  float* vnorm = wf + 12288;     // 16x128              (2048)
  float* dotb  = vnorm + 2048;   // 16x128              (2048)
  float* h2    = dotb + 2048;    // 16x128              (2048)
  float* u     = h2 + 2048;      // 16x384              (6144)  => 32768 floats

  const int block0 = blockIdx.x * 16;
  const int tid = threadIdx.x;
  const int wave = tid >> 5;
  const int lane = tid & 31;

  // phase 1: s1 = (s+ds)*eps ; v1 = (v+dv)*eps
  for (int i = tid; i < 2048; i += 256) {
    int nl = i >> 7, h = i & 127;
    int node = block0 + nl; if (node >= N) node = N - 1;
    size_t g = (size_t)node * 128 + h;
    s1[i] = (s[g] + ds[g]) * EPSF;
  }
  for (int i = tid; i < 6144; i += 256) {
    int row = i >> 7, h = i & 127;
    int d = row >> 4, nl = row & 15;
    int node = block0 + nl; if (node >= N) node = N - 1;
    size_t g = (size_t)node * 384 + (size_t)d * 128 + h;
    v1[i] = (v[g] + dv[g]) * EPSF;
  }
  __syncthreads();

  // phase 2: wf = v1 @ Wvm   (48 rows x 256 cols) -> 48 tiles, 6 per wave
  for (int t = wave; t < 48; t += 8) {
    const int mt = t >> 4;          // direction tile 0..2
    const int n0 = (t & 15) * 16;   // column tile
    v8f acc = {};
    acc = wmma_k(v1 + mt * 16 * HDIM, HDIM, Wvm + n0, 256, HDIM, lane, acc);
    store_tile(wf + (mt * 16) * 256 + n0, 256, acc, lane);
  }
  __syncthreads();

  // phase 3: vnorm = sqrt(sum_d v_r^2 + 1e-8), dot = sum_d v_r*v_l
  for (int i = tid; i < 2048; i += 256) {
    int nl = i >> 7, h = i & 127;
    float nrm = 1e-8f, dt = 0.0f;
#pragma unroll
    for (int d = 0; d < 3; ++d) {
      float vl = wf[(d * 16 + nl) * 256 + h];
      float vr = wf[(d * 16 + nl) * 256 + 128 + h];
      nrm += vr * vr;
      dt  += vr * vl;
    }
    vnorm[i] = sqrtf(nrm);
    dotb[i]  = dt;
  }
  __syncthreads();

  // phase 4: h2 = silu([s1|vnorm] @ Wm1 + bm1)   K=256 split into two halves
  {
    const int n0 = wave * 16;
    v8f acc = {};
    acc = wmma_k(s1,    HDIM, Wm1 + n0,              HDIM, HDIM, lane, acc);
    acc = wmma_k(vnorm, HDIM, Wm1 + 128 * HDIM + n0, HDIM, HDIM, lane, acc);
    const int row0 = (lane >> 4) << 3;
    const int l = lane & 15;
    const float bias = bm1[n0 + l];
#pragma unroll
    for (int vg = 0; vg < 8; ++vg)
      h2[(row0 + vg) * HDIM + n0 + l] = silu(acc[vg] + bias);
  }
  __syncthreads();

  // phase 5: u = h2 @ Wm2 + bm2   (16x384) -> 24 tiles, 3 per wave
  for (int t = wave; t < 24; t += 8) {
    const int n0 = t * 16;
    v8f acc = {};
    acc = wmma_k(h2, HDIM, Wm2 + n0, 3 * HDIM, HDIM, lane, acc);
    const int row0 = (lane >> 4) << 3;
    const int l = lane & 15;
    const float bias = bm2[n0 + l];
#pragma unroll
    for (int vg = 0; vg < 8; ++vg)
      u[(row0 + vg) * 384 + n0 + l] = acc[vg] + bias;
  }
  __syncthreads();

  // phase 6: combine and write packed output (N, 4, 128); out is write-once
  // -> non-temporal stores (don't evict L2-resident working sets)
  for (int i = tid; i < 2048; i += 256) {
    int nl = i >> 7, h = i & 127;
    int node = block0 + nl;
    if (node >= N) continue;
    float so = (s1[i] + u[nl * 384 + h] + u[nl * 384 + 256 + h] * dotb[i]) * EPSF;
    __builtin_nontemporal_store(so, out + (size_t)node * 512 + h);
#pragma unroll
    for (int d = 0; d < 3; ++d) {
      float vo = (v1[(d * 16 + nl) * 128 + h] +
                  wf[(d * 16 + nl) * 256 + h] * u[nl * 384 + 128 + h]) * EPSF;
      __builtin_nontemporal_store(vo, out + (size_t)node * 512 + (size_t)(1 + d) * 128 + h);
    }
  }
}

// ---------------------------------------------------------------------------
extern "C" void kernel_launch(void* const* d_in, const int* in_sizes, int n_in,
                              void* d_out, int out_size, void* d_ws, size_t ws_size,
                              hipStream_t stream) {
  const float* s        = (const float*)d_in[0];
  const float* v        = (const float*)d_in[1];
  const float* dir      = (const float*)d_in[2];
  const float* Wij      = (const float*)d_in[3];
  const int*   senders  = (const int*)d_in[4];
  const int*   receivers= (const int*)d_in[5];
  const float* Wi1      = (const float*)d_in[6];
  const float* bi1      = (const float*)d_in[7];
  const float* Wi2      = (const float*)d_in[8];
  const float* bi2      = (const float*)d_in[9];
  const float* Wm1      = (const float*)d_in[10];
  const float* bm1      = (const float*)d_in[11];
  const float* Wm2      = (const float*)d_in[12];
  const float* bm2      = (const float*)d_in[13];
  const float* Wvm      = (const float*)d_in[14];

  const int N = in_sizes[0] / HDIM;   // 50000
  const int E = in_sizes[4];          // 600000

  // workspace layout: x (N*384) | ds (N*128) | dv (N*384)
  float* ws = (float*)d_ws;
  float* x  = ws;
  float* ds = x + (size_t)N * 384;
  float* dv = ds + (size_t)N * 128;

  // zero the accumulators (ds|dv contiguous)
  (void)hipMemsetAsync(ds, 0, (size_t)N * 512 * sizeof(float), stream);

  const int nb = (N + 15) / 16;
  painn_node_mlp<<<nb, 256, 0, stream>>>(s, Wi1, bi1, Wi2, bi2, x, N);
  painn_edge<<<(E + 7) / 8, 256, 0, stream>>>(Wij, dir, senders, receivers,
                                              x, v, ds, dv, E);

  const int lds_bytes = 32768 * (int)sizeof(float);  // 128 KB
  (void)hipFuncSetAttribute((const void*)painn_update,
                            hipFuncAttributeMaxDynamicSharedMemorySize, lds_bytes);
  painn_update<<<nb, 256, lds_bytes, stream>>>(s, v, ds, dv, Wvm,
                                               Wm1, bm1, Wm2, bm2,
                                               (float*)d_out, N);
}